// Net_53764400611947
// MI455X (gfx1250) — compile-verified
//
#include <hip/hip_runtime.h>
#include <hip/hip_bf16.h>

typedef __attribute__((ext_vector_type(2))) float v2f;
typedef __attribute__((ext_vector_type(8))) float v8f;

#define FEAT 30
#define STRIDE 32   // padded row stride (128B per node row)

// ---------------- degree / normalization ----------------

__global__ void deg_init_kernel(float* deg, int n) {
  int i = blockIdx.x * blockDim.x + threadIdx.x;
  if (i < n) deg[i] = 1.0f;            // self-loop contribution
}

__global__ void deg_edge_kernel(const long long* __restrict__ dst, float* deg, int n_edges) {
  int i = blockIdx.x * blockDim.x + threadIdx.x;
  if (i < n_edges) atomicAdd(&deg[(int)dst[i]], 1.0f);
}

__global__ void deg_rsqrt_kernel(float* deg, int n) {
  int i = blockIdx.x * blockDim.x + threadIdx.x;
  if (i < n) deg[i] = rsqrtf(deg[i]);  // deg >= 1 guaranteed by self-loops
}

// pad x [N,FEAT] -> xp [N,STRIDE], zero pad columns (so GEMM loads are unconditional)
__global__ void pad_x_kernel(const float* __restrict__ x, float* __restrict__ xp, int n_nodes) {
  int i = blockIdx.x * blockDim.x + threadIdx.x;
  int node = i >> 5, c = i & 31;
  if (node >= n_nodes) return;
  xp[(size_t)node * STRIDE + c] = (c < FEAT) ? x[(size_t)node * FEAT + c] : 0.0f;
}

// ---------------- dense part: T = X @ W via V_WMMA_F32_16X16X4_F32 ----------------
// X is stride-32, zero-padded. W (30x30 row-major [k][n]) is staged transposed+padded
// in LDS as Wt[n*32+k] so both operand fetches are unconditional 8-byte loads.
// One wave32 per 16-node tile; each wave computes both 16-col tiles (acc0, acc1).
// A f32 16x4 layout: lanes 0-15 = rows M, VGPR0/1 = K=k0,k0+1 ; lanes 16-31 = K=k0+2,k0+3.
// B f32 4x16: lanes = N cols, same K split. C/D: VGPR v -> M=v (lanes 0-15) / M=v+8.
__global__ void __launch_bounds__(256)
gemm_wmma_kernel(const float* __restrict__ X, const float* __restrict__ W,
                 float* __restrict__ T, int n_nodes) {
  __shared__ float Wt[STRIDE * STRIDE];          // Wt[n*32 + k], zero-padded
  for (int i = threadIdx.x; i < STRIDE * STRIDE; i += blockDim.x) {
    int n = i >> 5, k = i & 31;
    Wt[i] = (n < FEAT && k < FEAT) ? W[k * FEAT + n] : 0.0f;
  }
  __syncthreads();

  const int wave = blockIdx.x * (blockDim.x >> 5) + (threadIdx.x >> 5);
  const int lane = threadIdx.x & 31;
  const int tiles = (n_nodes + 15) >> 4;
  if (wave >= tiles) return;                     // uniform per wave; no later syncthreads

  const int half = lane >> 4;
  const int l15  = lane & 15;
  const int row  = wave * 16 + l15;              // A: M for this lane (clamped below)
  const int rowc = row < n_nodes ? row : n_nodes - 1;
  const float* __restrict__ arow = X + (size_t)rowc * STRIDE;
  const float* __restrict__ b0p  = &Wt[(l15)      * STRIDE];  // col tile 0: N = l15
  const float* __restrict__ b1p  = &Wt[(l15 + 16) * STRIDE];  // col tile 1: N = l15+16

  v8f acc0 = {}, acc1 = {};
#pragma unroll
  for (int k0 = 0; k0 < 32; k0 += 4) {
    const int ka = k0 + half * 2;                // even -> 8B aligned
    v2f a  = *(const v2f*)(arow + ka);           // global_load_b64, unconditional
    v2f b0 = *(const v2f*)(b0p + ka);            // ds_load_b64, unconditional
    v2f b1 = *(const v2f*)(b1p + ka);
    acc0 = __builtin_amdgcn_wmma_f32_16x16x4_f32(false, a, false, b0, (short)0, acc0, false, false);
    acc1 = __builtin_amdgcn_wmma_f32_16x16x4_f32(false, a, false, b1, (short)0, acc1, false, false);
  }

  // C/D layout: VGPR v: lanes 0-15 -> M=v, lanes 16-31 -> M=v+8; N = lane&15
  const int mbase = wave * 16 + half * 8;
  if (wave * 16 + 16 <= n_nodes) {               // uniform fast path (always for N%16==0)
#pragma unroll
    for (int v = 0; v < 8; ++v) {
      float* trow = T + (size_t)(mbase + v) * STRIDE;
      trow[l15]      = acc0[v];
      trow[l15 + 16] = acc1[v];
    }
  } else {
#pragma unroll
    for (int v = 0; v < 8; ++v) {
      const int m = mbase + v;
      if (m < n_nodes) {
        float* trow = T + (size_t)m * STRIDE;
        trow[l15]      = acc0[v];
        trow[l15 + 16] = acc1[v];
      }
    }
  }
}

// ---------------- sparse part: agg = D^-1/2 (A+I) D^-1/2 applied to T ----------------

// self-loop term initializes the output buffer fully (also zeroes the pad lanes)
__global__ void agg_init_kernel(const float* __restrict__ T, const float* __restrict__ dinv,
                                float* __restrict__ agg, int n_nodes) {
  int i = blockIdx.x * blockDim.x + threadIdx.x;
  int node = i >> 5, c = i & 31;
  if (node >= n_nodes) return;
  float di = dinv[node];
  agg[(size_t)node * STRIDE + c] = (c < FEAT) ? T[(size_t)node * STRIDE + c] * di * di : 0.0f;
}

// one wave32 per edge; lane = feature channel; 128B-aligned gather + fp32 scatter atomics
__global__ void agg_edge_kernel(const float* __restrict__ T, const float* __restrict__ dinv,
                                const long long* __restrict__ srcIdx,
                                const long long* __restrict__ dstIdx,
                                float* __restrict__ agg, int n_edges) {
  const int edge = blockIdx.x * (blockDim.x >> 5) + (threadIdx.x >> 5);
  const int lane = threadIdx.x & 31;
  if (edge >= n_edges) return;
  if (lane == 0 && edge + 4096 < n_edges) {      // gfx1250 global_prefetch_b8
    __builtin_prefetch(&srcIdx[edge + 4096], 0, 1);
    __builtin_prefetch(&dstIdx[edge + 4096], 0, 1);
  }
  const int s = (int)srcIdx[edge];
  const int d = (int)dstIdx[edge];
  const float w = dinv[s] * dinv[d];
  if (lane < FEAT)
    atomicAdd(&agg[(size_t)d * STRIDE + lane], T[(size_t)s * STRIDE + lane] * w);
}

__global__ void bias_relu_kernel(float* __restrict__ h, const float* __restrict__ b, int n_nodes) {
  int i = blockIdx.x * blockDim.x + threadIdx.x;
  int node = i >> 5, c = i & 31;
  if (node >= n_nodes) return;
  float v = 0.0f;
  if (c < FEAT) {
    v = h[(size_t)node * STRIDE + c] + b[c];
    v = v > 0.0f ? v : 0.0f;
  }
  h[(size_t)node * STRIDE + c] = v;
}

// ---------------- pooling + linear head ----------------

__global__ void pool_zero_kernel(float* pooled, int n) {
  int i = blockIdx.x * blockDim.x + threadIdx.x;
  if (i < n) pooled[i] = 0.0f;
}

__global__ void pool_kernel(const float* __restrict__ h, const long long* __restrict__ bids,
                            float* __restrict__ pooled, int n_nodes) {
  const int node = blockIdx.x * (blockDim.x >> 5) + (threadIdx.x >> 5);
  const int lane = threadIdx.x & 31;
  if (node >= n_nodes) return;
  const int g = (int)bids[node];
  if (lane < FEAT)
    atomicAdd(&pooled[(size_t)g * STRIDE + lane], h[(size_t)node * STRIDE + lane]);
}

__global__ void head_kernel(const float* __restrict__ pooled, const float* __restrict__ linW,
                            const float* __restrict__ linb, float* __restrict__ out, int n_graphs) {
  const int g = blockIdx.x;
  const int lane = threadIdx.x;
  if (g >= n_graphs) return;
  float v = (lane < FEAT) ? pooled[(size_t)g * STRIDE + lane] * linW[lane] : 0.0f;
#pragma unroll
  for (int off = 16; off > 0; off >>= 1) v += __shfl_xor(v, off, 32);
  if (lane == 0) out[g] = v + linb[0];
}

// ---------------- launcher ----------------

extern "C" void kernel_launch(void* const* d_in, const int* in_sizes, int n_in,
                              void* d_out, int out_size, void* d_ws, size_t ws_size,
                              hipStream_t stream) {
  const float*     x    = (const float*)d_in[0];
  const long long* ei   = (const long long*)d_in[1];
  const long long* bids = (const long long*)d_in[2];
  const float* W1 = (const float*)d_in[3];
  const float* b1 = (const float*)d_in[4];
  const float* W2 = (const float*)d_in[5];
  const float* b2 = (const float*)d_in[6];
  const float* W3 = (const float*)d_in[7];
  const float* b3 = (const float*)d_in[8];
  const float* linW = (const float*)d_in[9];
  const float* linb = (const float*)d_in[10];
  float* out = (float*)d_out;

  const int n_nodes  = in_sizes[0] / FEAT;
  const int n_edges  = in_sizes[1] / 2;
  const int n_graphs = out_size;

  const long long* srcIdx = ei;            // edge_index[0]
  const long long* dstIdx = ei + n_edges;  // edge_index[1]

  // workspace layout (keep the big buffers 128B-aligned)
  float* wsf = (float*)d_ws;
  const size_t nAlign = ((size_t)n_nodes + 31) & ~(size_t)31;
  float* dinv   = wsf;                                   // n_nodes (padded to 32)
  float* bufA   = dinv + nAlign;                         // n_nodes*32 (GEMM out)
  float* bufB   = bufA + (size_t)n_nodes * STRIDE;       // n_nodes*32 (padded x / agg / h)
  float* pooled = bufB + (size_t)n_nodes * STRIDE;       // n_graphs*32

  const int TPB = 256;
  const int wpb = TPB / 32;

  const int nblk      = (n_nodes + TPB - 1) / TPB;
  const int eblk      = (n_edges + TPB - 1) / TPB;
  const int ew_blk    = (int)(((size_t)n_nodes * STRIDE + TPB - 1) / TPB); // elementwise
  const int edgew_blk = (n_edges + wpb - 1) / wpb;                         // 1 wave/edge
  const int tiles     = (n_nodes + 15) / 16;
  const int gemm_blk  = (tiles + wpb - 1) / wpb;                           // 1 wave/16-node tile
  const int poolw_blk = (n_nodes + wpb - 1) / wpb;

  // normalization: deg = 1 + indegree ; dinv = rsqrt(deg)
  deg_init_kernel<<<nblk, TPB, 0, stream>>>(dinv, n_nodes);
  deg_edge_kernel<<<eblk, TPB, 0, stream>>>(dstIdx, dinv, n_edges);
  deg_rsqrt_kernel<<<nblk, TPB, 0, stream>>>(dinv, n_nodes);

  // pad x into bufB (stride 32, zero pad cols)
  pad_x_kernel<<<ew_blk, TPB, 0, stream>>>(x, bufB, n_nodes);

  // layer 1: bufB @ W1 -> bufA ; aggregate -> bufB ; +b1, relu
  gemm_wmma_kernel<<<gemm_blk, TPB, 0, stream>>>(bufB, W1, bufA, n_nodes);
  agg_init_kernel<<<ew_blk, TPB, 0, stream>>>(bufA, dinv, bufB, n_nodes);
  agg_edge_kernel<<<edgew_blk, TPB, 0, stream>>>(bufA, dinv, srcIdx, dstIdx, bufB, n_edges);
  bias_relu_kernel<<<ew_blk, TPB, 0, stream>>>(bufB, b1, n_nodes);

  // layer 2
  gemm_wmma_kernel<<<gemm_blk, TPB, 0, stream>>>(bufB, W2, bufA, n_nodes);
  agg_init_kernel<<<ew_blk, TPB, 0, stream>>>(bufA, dinv, bufB, n_nodes);
  agg_edge_kernel<<<edgew_blk, TPB, 0, stream>>>(bufA, dinv, srcIdx, dstIdx, bufB, n_edges);
  bias_relu_kernel<<<ew_blk, TPB, 0, stream>>>(bufB, b2, n_nodes);

  // layer 3
  gemm_wmma_kernel<<<gemm_blk, TPB, 0, stream>>>(bufB, W3, bufA, n_nodes);
  agg_init_kernel<<<ew_blk, TPB, 0, stream>>>(bufA, dinv, bufB, n_nodes);
  agg_edge_kernel<<<edgew_blk, TPB, 0, stream>>>(bufA, dinv, srcIdx, dstIdx, bufB, n_edges);
  bias_relu_kernel<<<ew_blk, TPB, 0, stream>>>(bufB, b3, n_nodes);

  // global add pool + linear head
  pool_zero_kernel<<<(n_graphs * STRIDE + TPB - 1) / TPB, TPB, 0, stream>>>(pooled, n_graphs * STRIDE);
  pool_kernel<<<poolw_blk, TPB, 0, stream>>>(bufB, bids, pooled, n_nodes);
  head_kernel<<<n_graphs, 32, 0, stream>>>(pooled, linW, linb, out, n_graphs);
}